// DifferentiableDAG_73452530696627
// MI455X (gfx1250) — compile-verified
//
#include <hip/hip_runtime.h>
#include <hip/hip_bf16.h>
#include <math.h>
#include <stdint.h>

// Sizes from the reference
#define B_     2048
#define H_     1024
#define N0_    4
#define STEPS_ 8
#define NMAX_  12      // N0 + STEPS
#define NOPS_  9
#define MLPK_  (4*H_ + NOPS_)   // 4105, row stride of Wmlp

typedef float v2f __attribute__((ext_vector_type(2)));
typedef float v8f __attribute__((ext_vector_type(8)));

// ---------------------------------------------------------------------------
// Threefry2x32-20 (JAX PRNG) — key derivation + gumbel sampling
// ---------------------------------------------------------------------------
__device__ __forceinline__ uint32_t rotl32(uint32_t v, int s) {
  return (v << s) | (v >> (32 - s));
}

__device__ __forceinline__ void tf2x32(uint32_t k0, uint32_t k1,
                                       uint32_t c0, uint32_t c1,
                                       uint32_t& o0, uint32_t& o1) {
  const uint32_t ks0 = k0, ks1 = k1, ks2 = k0 ^ k1 ^ 0x1BD11BDAu;
  uint32_t x0 = c0 + ks0, x1 = c1 + ks1;
  const int rA[4] = {13, 15, 26, 6};
  const int rB[4] = {17, 29, 16, 24};
#pragma unroll
  for (int i = 0; i < 4; i++) { x0 += x1; x1 = rotl32(x1, rA[i]); x1 ^= x0; }
  x0 += ks1; x1 += ks2 + 1u;
#pragma unroll
  for (int i = 0; i < 4; i++) { x0 += x1; x1 = rotl32(x1, rB[i]); x1 ^= x0; }
  x0 += ks2; x1 += ks0 + 2u;
#pragma unroll
  for (int i = 0; i < 4; i++) { x0 += x1; x1 = rotl32(x1, rA[i]); x1 ^= x0; }
  x0 += ks0; x1 += ks1 + 3u;
#pragma unroll
  for (int i = 0; i < 4; i++) { x0 += x1; x1 = rotl32(x1, rB[i]); x1 ^= x0; }
  x0 += ks1; x1 += ks2 + 4u;
#pragma unroll
  for (int i = 0; i < 4; i++) { x0 += x1; x1 = rotl32(x1, rA[i]); x1 ^= x0; }
  x0 += ks2; x1 += ks0 + 5u;
  o0 = x0; o1 = x1;
}

// kk = {k1.0,k1.1, k2.0,k2.1, k3.0,k3.1} for step: split(fold_in(key(1234),step),3)
__device__ __forceinline__ void step_keys(int step, uint32_t* kk) {
  uint32_t f0, f1;
  tf2x32(0u, 1234u, 0u, (uint32_t)step, f0, f1);           // fold_in
  uint32_t a0, a1, b0, b1, c0, c1;                          // split -> bits over iota(6)
  tf2x32(f0, f1, 0u, 3u, a0, a1);
  tf2x32(f0, f1, 1u, 4u, b0, b1);
  tf2x32(f0, f1, 2u, 5u, c0, c1);
  // out = [a0,b0,c0,a1,b1,c1]; keys are consecutive pairs
  kk[0] = a0; kk[1] = b0;   // k1
  kk[2] = c0; kk[3] = a1;   // k2
  kk[4] = b1; kk[5] = c1;   // k3
}

// gumbel sample for flat element i of an n-element tensor (n even)
__device__ __forceinline__ float gumbel_at(uint32_t k0, uint32_t k1,
                                           uint32_t i, uint32_t n) {
  const uint32_t h = n >> 1;
  uint32_t o0, o1, bits;
  if (i < h) { tf2x32(k0, k1, i, i + h, o0, o1); bits = o0; }
  else       { tf2x32(k0, k1, i - h, i, o0, o1); bits = o1; }
  const uint32_t fb = (bits >> 9) | 0x3f800000u;
  float f = __uint_as_float(fb) - 1.0f;       // [0,1)
  const float tiny = 1.1754943508e-38f;
  float u = tiny + f * (1.0f - tiny);         // [tiny,1)
  return -logf(-logf(u));
}

__device__ __forceinline__ float clamp40(float x) {
  return fminf(fmaxf(x, -40.0f), 40.0f);
}

// ---------------------------------------------------------------------------
// Block reduction (wave32-aware, 256 threads = 8 waves)
// ---------------------------------------------------------------------------
__device__ __forceinline__ float block_reduce(float v, float* sh) {
#pragma unroll
  for (int off = 16; off > 0; off >>= 1) v += __shfl_xor(v, off, 32);
  const int w = threadIdx.x >> 5;
  if ((threadIdx.x & 31) == 0) sh[w] = v;
  __syncthreads();
  float r = (threadIdx.x < 8) ? sh[threadIdx.x] : 0.0f;
#pragma unroll
  for (int off = 4; off > 0; off >>= 1) r += __shfl_xor(r, off, 32);
  if (threadIdx.x == 0) sh[8] = r;
  __syncthreads();
  return sh[8];
}

// ---------------------------------------------------------------------------
// WMMA fp32 GEMM: C[M,N] = A[M,K] * B + (bias)
//   TRANS_B=1: B[k][j] = Bw[j*ldb + k]   (weight stored [N_out, K] row-major)
//   TRANS_B=0: B[k][j] = Bw[k*ldb + j]
// Block = 256 threads (8 waves); tile 64(M) x 128(N) x 16(K), double-buffered
// LDS. Each wave: 16x64 strip = 4 x v_wmma_f32_16x16x4_f32 accumulators.
//
// LDS tiles are stored PAIR-INTERLEAVED: element (k, x) lives at
// [k>>1][x][k&1], so a WMMA fragment {v(k), v(k+1)} is one aligned b64 load
// into an even VGPR pair (no fix-up moves). The inner dim is padded by 16 so
// the k-parity stride is 128B (mod 256B): lanes 16-31 (which read k+2 at the
// same x) land on the other 32 LDS banks -> conflict-free wave-wide b64 reads.
// Requires M%64==0, N%128==0, K%16==0 (true for all uses here).
// ---------------------------------------------------------------------------
#define AP_ 80    // 64 + 16 pad
#define BP_ 144   // 128 + 16 pad

template <int TRANS_B>
__global__ __launch_bounds__(256) void gemm_wmma(
    const float* __restrict__ A, int lda,
    const float* __restrict__ Bw, int ldb,
    const float* __restrict__ bias,
    float* __restrict__ C, int ldc, int K) {
  __shared__ float As[2][8][AP_][2];    // [k/2][m][k&1]
  __shared__ float Bs[2][8][BP_][2];    // [k/2][n][k&1]

  const int tid  = threadIdx.x;
  const int lane = tid & 31;
  const int l15  = lane & 15;
  const int hi2  = (lane >> 4) << 1;             // 0 for lanes 0-15, 2 for 16-31
  const int wave = tid >> 5;
  const int m0   = (wave & 3) * 16;
  const int n0   = (wave >> 2) * 64;
  const int mbase = blockIdx.y * 64;
  const int nbase = blockIdx.x * 128;

  v8f acc[4];
  const v8f vzero = {};
#pragma unroll
  for (int t = 0; t < 4; t++) acc[t] = vzero;

  auto load_tile = [&](int buf, int kt) {
    {
      const int row = tid >> 2, k0 = (tid & 3) * 4;
      const float4 va =
          *(const float4*)(A + (size_t)(mbase + row) * lda + kt + k0);
      const v2f lo = {va.x, va.y}, hi = {va.z, va.w};
      *(v2f*)&As[buf][(k0 >> 1) + 0][row][0] = lo;
      *(v2f*)&As[buf][(k0 >> 1) + 1][row][0] = hi;
    }
    if (TRANS_B) {   // contiguous along k in memory (ldb may be odd: scalar ld)
#pragma unroll
      for (int r = 0; r < 2; r++) {
        const int lin = tid + 256 * r;
        const int j = lin >> 2, k0 = (lin & 3) * 4;
        const float* src = Bw + (size_t)(nbase + j) * ldb + kt + k0;
        const v2f lo = {src[0], src[1]}, hi = {src[2], src[3]};
        *(v2f*)&Bs[buf][(k0 >> 1) + 0][j][0] = lo;
        *(v2f*)&Bs[buf][(k0 >> 1) + 1][j][0] = hi;
      }
    } else {         // contiguous along j in memory (ldb % 4 == 0 here)
#pragma unroll
      for (int r = 0; r < 2; r++) {
        const int lin = tid + 256 * r;
        const int kr = lin >> 5, j0 = (lin & 31) * 4;
        const float4 vb =
            *(const float4*)(Bw + (size_t)(kt + kr) * ldb + nbase + j0);
        Bs[buf][kr >> 1][j0 + 0][kr & 1] = vb.x;
        Bs[buf][kr >> 1][j0 + 1][kr & 1] = vb.y;
        Bs[buf][kr >> 1][j0 + 2][kr & 1] = vb.z;
        Bs[buf][kr >> 1][j0 + 3][kr & 1] = vb.w;
      }
    }
  };

  auto afrag = [&](int buf, int kk) -> v2f {
    return *(const v2f*)&As[buf][(kk + hi2) >> 1][m0 + l15][0];
  };
  auto bfrag = [&](int buf, int kk, int t) -> v2f {
    return *(const v2f*)&Bs[buf][(kk + hi2) >> 1][n0 + 16 * t + l15][0];
  };

  load_tile(0, 0);
  __syncthreads();

  // register double-buffered fragments
  v2f af[2];
  v2f bfr[2][4];
  af[0] = afrag(0, 0);
#pragma unroll
  for (int t = 0; t < 4; t++) bfr[0][t] = bfrag(0, 0, t);

  int buf = 0;
  for (int kt = 0; kt < K; kt += 16) {
    if (kt + 16 < K) load_tile(buf ^ 1, kt + 16);
    if (kt + 32 < K)  // hint tile-after-next into cache hierarchy
      __builtin_prefetch(A + (size_t)(mbase + (tid >> 2)) * lda + kt + 32, 0, 0);
#pragma unroll
    for (int ki = 0; ki < 4; ki++) {
      const int cur = ki & 1, nxt = cur ^ 1;
      if (ki < 3) {   // prefetch next kk-step's fragments while WMMAs issue
        af[nxt] = afrag(buf, 4 * ki + 4);
#pragma unroll
        for (int t = 0; t < 4; t++) bfr[nxt][t] = bfrag(buf, 4 * ki + 4, t);
      }
#pragma unroll
      for (int t = 0; t < 4; t++)
        acc[t] = __builtin_amdgcn_wmma_f32_16x16x4_f32(
            false, af[cur], false, bfr[cur][t], (short)0, acc[t], false, false);
    }
    __syncthreads();
    buf ^= 1;
    if (kt + 16 < K) {   // first fragments of the freshly filled buffer
      af[0] = afrag(buf, 0);
#pragma unroll
      for (int t = 0; t < 4; t++) bfr[0][t] = bfrag(buf, 0, t);
    }
  }

  // C layout: VGPR r -> M = r (+8 for lanes 16-31), N = lane&15
  const int rowb = mbase + m0 + ((lane >> 4) ? 8 : 0);
#pragma unroll
  for (int t = 0; t < 4; t++) {
    const int col = nbase + n0 + 16 * t + l15;
    const float bv = bias ? bias[col] : 0.0f;
#pragma unroll
    for (int r = 0; r < 8; r++)
      C[(size_t)(rowb + r) * ldc + col] = acc[t][r] + bv;
  }
}

// ---------------------------------------------------------------------------
// Init: embeds0 [B,4,H] -> emb [B,12,H]; values0 [B,4] -> val [B,12]
// ---------------------------------------------------------------------------
__global__ void init_kernel(const float* __restrict__ e0,
                            const float* __restrict__ v0,
                            float* __restrict__ emb, float* __restrict__ val) {
  const int idx = blockIdx.x * blockDim.x + threadIdx.x;
  const int tot = B_ * N0_ * H_;
  if (idx < tot) {
    const int h = idx & (H_ - 1);
    const int bn = idx >> 10;
    const int n = bn % N0_, b = bn / N0_;
    emb[((size_t)b * NMAX_ + n) * H_ + h] = e0[idx];
  }
  if (idx < B_ * N0_) {
    const int n = idx % N0_, b = idx / N0_;
    val[b * NMAX_ + n] = v0[idx];
  }
}

// cw1 = Wmlp[:,2H:3H]@w_s2e ; cw2 = Wmlp[:,3H:4H]@w_s2e ;
// cb  = Wmlp[:,2H:3H]@b_s2e + Wmlp[:,3H:4H]@b_s2e + bmlp
__global__ void precomp_kernel(const float* __restrict__ Wmlp,
                               const float* __restrict__ bmlp,
                               const float* __restrict__ wse,
                               const float* __restrict__ bse,
                               float* __restrict__ cw1, float* __restrict__ cw2,
                               float* __restrict__ cb) {
  const int j = blockIdx.x * blockDim.x + threadIdx.x;
  if (j >= H_) return;
  const float* row = Wmlp + (size_t)j * MLPK_;
  float a = 0, b = 0, c = 0, d = 0;
  for (int k = 0; k < H_; k++) {
    const float w3 = row[2 * H_ + k], w4 = row[3 * H_ + k];
    a += w3 * wse[k]; b += w4 * wse[k];
    c += w3 * bse[k]; d += w4 * bse[k];
  }
  cw1[j] = a; cw2[j] = b; cb[j] = c + d + bmlp[j];
}

__global__ void ctx_kernel(const float* __restrict__ opc,
                           const float* __restrict__ ppc,
                           const float* __restrict__ stepe, int step,
                           float* __restrict__ octx, float* __restrict__ pctx) {
  const int idx = blockIdx.x * blockDim.x + threadIdx.x;
  if (idx >= B_ * H_) return;
  const float e = stepe[step * H_ + (idx & (H_ - 1))];
  octx[idx] = opc[idx] + e;
  pctx[idx] = ppc[idx] + e;
}

// ---------------------------------------------------------------------------
// Attention select: s = clip(r.e + q.bk); att = onehot(argmax(s+g)) + 1e-10
// e12[b] = [e1,e2]; v1/v2 scalars. One 256-thread block per batch row.
// ---------------------------------------------------------------------------
__global__ __launch_bounds__(256) void attn_kernel(
    const float* __restrict__ r1, const float* __restrict__ r2,
    const float* __restrict__ q1, const float* __restrict__ q2,
    const float* __restrict__ bk, const float* __restrict__ emb,
    const float* __restrict__ val, float* __restrict__ e12,
    float* __restrict__ v1o, float* __restrict__ v2o, int N, int step) {
  __shared__ float sh[16];
  __shared__ int sa1, sa2;
  const int b = blockIdx.x, tid = threadIdx.x;
  const float* r1b = r1 + (size_t)b * H_;
  const float* r2b = r2 + (size_t)b * H_;
  const float* q1b = q1 + (size_t)b * H_;
  const float* q2b = q2 + (size_t)b * H_;

  float pq1 = 0, pq2 = 0;
  float p1[NMAX_ - 1], p2[NMAX_ - 1];
#pragma unroll
  for (int n = 0; n < NMAX_ - 1; n++) { p1[n] = 0; p2[n] = 0; }

  for (int h = tid; h < H_; h += 256) {
    const float r1v = r1b[h], r2v = r2b[h];
    pq1 += q1b[h] * bk[h];
    pq2 += q2b[h] * bk[h];
#pragma unroll
    for (int n = 0; n < NMAX_ - 1; n++)
      if (n < N) {
        const float ev = emb[((size_t)b * NMAX_ + n) * H_ + h];
        p1[n] += r1v * ev;
        p2[n] += r2v * ev;
      }
  }
  const float qb1 = block_reduce(pq1, sh);
  const float qb2 = block_reduce(pq2, sh);
  float ls1[NMAX_ - 1], ls2[NMAX_ - 1];
#pragma unroll
  for (int n = 0; n < NMAX_ - 1; n++)
    if (n < N) {
      ls1[n] = block_reduce(p1[n], sh);
      ls2[n] = block_reduce(p2[n], sh);
    }

  if (tid == 0) {
    uint32_t kk[6];
    step_keys(step, kk);
    const uint32_t ntot = (uint32_t)(B_ * N);
    float best1 = -3.4e38f, best2 = -3.4e38f;
    int a1 = 0, a2 = 0;
#pragma unroll
    for (int n = 0; n < NMAX_ - 1; n++)
      if (n < N) {
        const uint32_t i = (uint32_t)(b * N + n);
        const float z1 = clamp40(ls1[n] + qb1) + gumbel_at(kk[0], kk[1], i, ntot);
        const float z2 = clamp40(ls2[n] + qb2) + gumbel_at(kk[2], kk[3], i, ntot);
        if (z1 > best1) { best1 = z1; a1 = n; }
        if (z2 > best2) { best2 = z2; a2 = n; }
      }
    sa1 = a1; sa2 = a2;
    float sv = 0;
    for (int n = 0; n < N; n++) sv += val[b * NMAX_ + n];
    v1o[b] = val[b * NMAX_ + a1] + 1e-10f * sv;
    v2o[b] = val[b * NMAX_ + a2] + 1e-10f * sv;
  }
  __syncthreads();
  const int a1 = sa1, a2 = sa2;
  for (int h = tid; h < H_; h += 256) {
    float se = 0;
#pragma unroll
    for (int n = 0; n < NMAX_ - 1; n++)
      if (n < N) se += emb[((size_t)b * NMAX_ + n) * H_ + h];
    e12[(size_t)b * 2 * H_ + h] =
        emb[((size_t)b * NMAX_ + a1) * H_ + h] + 1e-10f * se;
    e12[(size_t)b * 2 * H_ + H_ + h] =
        emb[((size_t)b * NMAX_ + a2) * H_ + h] + 1e-10f * se;
  }
}

// ---------------------------------------------------------------------------
// Op select: logits = clip(opq@Wsel^T + bsel); op_w = onehot(argmax(+g)) + 1e-10
// new_val = outs[amax] + 1e-10*sum(outs). One block per batch row.
// ---------------------------------------------------------------------------
__global__ __launch_bounds__(256) void op_kernel(
    const float* __restrict__ opq, const float* __restrict__ Wsel,
    const float* __restrict__ bsel, const float* __restrict__ v1i,
    const float* __restrict__ v2i, float* __restrict__ opw,
    float* __restrict__ val, int step) {
  __shared__ float sh[16];
  const int b = blockIdx.x, tid = threadIdx.x;
  const float* q = opq + (size_t)b * H_;
  float po[NOPS_];
#pragma unroll
  for (int o = 0; o < NOPS_; o++) po[o] = 0;
  for (int h = tid; h < H_; h += 256) {
    const float qv = q[h];
#pragma unroll
    for (int o = 0; o < NOPS_; o++) po[o] += qv * Wsel[o * H_ + h];
  }
  float lg[NOPS_];
#pragma unroll
  for (int o = 0; o < NOPS_; o++)
    lg[o] = clamp40(block_reduce(po[o], sh) + bsel[o]);

  if (tid == 0) {
    uint32_t kk[6];
    step_keys(step, kk);
    const uint32_t ntot = (uint32_t)(B_ * NOPS_);
    float best = -3.4e38f;
    int amax = 0;
#pragma unroll
    for (int o = 0; o < NOPS_; o++) {
      const float z =
          lg[o] + gumbel_at(kk[4], kk[5], (uint32_t)(b * NOPS_ + o), ntot);
      if (z > best) { best = z; amax = o; }
    }
    const float v1 = v1i[b], v2 = v2i[b];
    const float dv = fminf(fmaxf(v1 / (fabsf(v2) + 1e-8f), -1e6f), 1e6f);
    const float xs = fminf(fmaxf(fabsf(v1) + 1e-8f, 1e-8f), 1e3f);
    const float ys = fminf(fmaxf(v2, -6.0f), 6.0f);
    const float sg = (v1 > 0.f) ? 1.f : ((v1 < 0.f) ? -1.f : 0.f);
    const float pw = fminf(fmaxf(powf(xs, ys) * sg, -1e6f), 1e6f);
    const float lgo =
        fminf(fmaxf(logf(fmaxf(fabsf(v1) + 1e-8f, 1e-8f)), -20.f), 20.f);
    float outs[NOPS_] = {v1 + v2, v1, v1 * v2, v1 - v2, dv, pw, lgo,
                         fmaxf(v1, v2), fminf(v1, v2)};
    float so = 0;
#pragma unroll
    for (int o = 0; o < NOPS_; o++) so += outs[o];
    val[b * NMAX_ + N0_ + step] = outs[amax] + 1e-10f * so;
#pragma unroll
    for (int o = 0; o < NOPS_; o++)
      opw[b * 16 + o] = (o == amax ? 1.0f : 0.0f) + 1e-10f;
  }
}

// new_emb = 0.5*(e1+e2) + delta + v1*cw1 + v2*cw2 + cb + op_w @ Wmlp[:,4H:]^T
__global__ void fin_kernel(const float* __restrict__ delta,
                           const float* __restrict__ e12,
                           const float* __restrict__ v1,
                           const float* __restrict__ v2,
                           const float* __restrict__ cw1,
                           const float* __restrict__ cw2,
                           const float* __restrict__ cb,
                           const float* __restrict__ opw,
                           const float* __restrict__ Wmlp,
                           float* __restrict__ emb, int step) {
  const int idx = blockIdx.x * blockDim.x + threadIdx.x;
  if (idx >= B_ * H_) return;
  const int j = idx & (H_ - 1);
  const int b = idx >> 10;
  const float e1 = e12[(size_t)b * 2 * H_ + j];
  const float e2 = e12[(size_t)b * 2 * H_ + H_ + j];
  float a = 0.5f * (e1 + e2) + delta[idx] + v1[b] * cw1[j] + v2[b] * cw2[j] +
            cb[j];
  const float* wrow = Wmlp + (size_t)j * MLPK_ + 4 * H_;
#pragma unroll
  for (int o = 0; o < NOPS_; o++) a += opw[b * 16 + o] * wrow[o];
  emb[((size_t)b * NMAX_ + N0_ + step) * H_ + j] = a;
}

// out[b,n,0:H] = emb[b,n,:]; out[b,n,H] = val[b,n]
__global__ void pack_kernel(const float* __restrict__ emb,
                            const float* __restrict__ val,
                            float* __restrict__ out) {
  const size_t idx = (size_t)blockIdx.x * blockDim.x + threadIdx.x;
  const size_t total = (size_t)B_ * NMAX_ * (H_ + 1);
  if (idx >= total) return;
  const int c = (int)(idx % (H_ + 1));
  const size_t bn = idx / (H_ + 1);
  out[idx] = (c < H_) ? emb[bn * H_ + c] : val[bn];
}

// ---------------------------------------------------------------------------
extern "C" void kernel_launch(void* const* d_in, const int* in_sizes, int n_in,
                              void* d_out, int out_size, void* d_ws,
                              size_t ws_size, hipStream_t stream) {
  const float* embeds0 = (const float*)d_in[0];
  const float* values0 = (const float*)d_in[1];
  const float* operand_ctx = (const float*)d_in[2];
  const float* op_ctx = (const float*)d_in[3];
  const float* Wq1 = (const float*)d_in[4];
  const float* bq1 = (const float*)d_in[5];
  const float* Wq2 = (const float*)d_in[6];
  const float* bq2 = (const float*)d_in[7];
  const float* Wopq = (const float*)d_in[8];
  const float* bopq = (const float*)d_in[9];
  const float* Wk = (const float*)d_in[10];
  const float* bk = (const float*)d_in[11];
  const float* Wsel = (const float*)d_in[12];
  const float* bsel = (const float*)d_in[13];
  const float* step_emb = (const float*)d_in[14];
  const float* Wmlp = (const float*)d_in[15];
  const float* bmlp = (const float*)d_in[16];
  const float* w_s2e = (const float*)d_in[17];
  const float* b_s2e = (const float*)d_in[18];
  float* out = (float*)d_out;

  float* ws = (float*)d_ws;
  size_t off = 0;
  auto alloc = [&](size_t n) { float* p = ws + off; off += n; return p; };
  const size_t BH = (size_t)B_ * H_;
  float* emb = alloc((size_t)B_ * NMAX_ * H_);
  float* val = alloc((size_t)B_ * NMAX_);
  float* octx = alloc(BH);
  float* pctx = alloc(BH);
  float* q1 = alloc(BH);
  float* q2 = alloc(BH);
  float* opq = alloc(BH);
  float* r1 = alloc(BH);
  float* r2 = alloc(BH);
  float* e12 = alloc(2 * BH);
  float* delta = alloc(BH);
  float* opw = alloc((size_t)B_ * 16);
  float* v1 = alloc(B_);
  float* v2 = alloc(B_);
  float* cw1 = alloc(H_);
  float* cw2 = alloc(H_);
  float* cb = alloc(H_);
  (void)in_sizes; (void)n_in; (void)out_size; (void)ws_size;

  const dim3 blk(256);
  init_kernel<<<dim3((B_ * N0_ * H_ + 255) / 256), blk, 0, stream>>>(
      embeds0, values0, emb, val);
  precomp_kernel<<<dim3((H_ + 255) / 256), blk, 0, stream>>>(
      Wmlp, bmlp, w_s2e, b_s2e, cw1, cw2, cb);

  const dim3 ggrid(H_ / 128, B_ / 64);  // N-tiles x M-tiles = 8 x 32
  for (int step = 0; step < STEPS_; step++) {
    const int N = N0_ + step;
    ctx_kernel<<<dim3((B_ * H_ + 255) / 256), blk, 0, stream>>>(
        operand_ctx, op_ctx, step_emb, step, octx, pctx);
    gemm_wmma<1><<<ggrid, blk, 0, stream>>>(octx, H_, Wq1, H_, bq1, q1, H_, H_);
    gemm_wmma<1><<<ggrid, blk, 0, stream>>>(octx, H_, Wq2, H_, bq2, q2, H_, H_);
    gemm_wmma<1><<<ggrid, blk, 0, stream>>>(pctx, H_, Wopq, H_, bopq, opq, H_, H_);
    gemm_wmma<0><<<ggrid, blk, 0, stream>>>(q1, H_, Wk, H_, nullptr, r1, H_, H_);
    gemm_wmma<0><<<ggrid, blk, 0, stream>>>(q2, H_, Wk, H_, nullptr, r2, H_, H_);
    attn_kernel<<<dim3(B_), blk, 0, stream>>>(r1, r2, q1, q2, bk, emb, val, e12,
                                              v1, v2, N, step);
    op_kernel<<<dim3(B_), blk, 0, stream>>>(opq, Wsel, bsel, v1, v2, opw, val,
                                            step);
    gemm_wmma<1><<<ggrid, blk, 0, stream>>>(e12, 2 * H_, Wmlp, MLPK_, nullptr,
                                            delta, H_, 2 * H_);
    fin_kernel<<<dim3((B_ * H_ + 255) / 256), blk, 0, stream>>>(
        delta, e12, v1, v2, cw1, cw2, cb, opw, Wmlp, emb, step);
  }
  const size_t total = (size_t)B_ * NMAX_ * (H_ + 1);
  pack_kernel<<<dim3((unsigned)((total + 255) / 256)), blk, 0, stream>>>(
      emb, val, out);
}